// PointRender_45543833207282
// MI455X (gfx1250) — compile-verified
//
#include <hip/hip_runtime.h>

#define EMBED 96
#define NCLS  3
#define HH    512
#define WW    512
#define HW    (HH * WW)            // 262144 = 1<<18
#define NPTS  16384
#define NIMP  12288
#define NCOV  4096
#define KN    49152                // K * N oversampled candidates per batch
#define PAD   65536                // bitonic pad (next pow2 above 49152)
#define BB    4
#define CHUNK 8192                 // elements per LDS sort chunk (64 KB of u64)
#define SORT_T 1024                // threads per sort block

typedef __attribute__((ext_vector_type(2)))  float    v2f;
typedef __attribute__((ext_vector_type(8)))  float    v8f;

__device__ __forceinline__ unsigned int float_flip(float f) {
  unsigned int u = __float_as_uint(f);
  return u ^ ((u >> 31) ? 0xFFFFFFFFu : 0x80000000u);
}

__device__ __forceinline__ int iclamp(int v, int lo, int hi) {
  return v < lo ? lo : (v > hi ? hi : v);
}

// ---------------------------------------------------------------------------
// Kernel 1: per-candidate uncertainty -> 64-bit sort key.
// key = (orderflip(uncert) << 32) | (0xFFFFFFFF - index). Descending sort of
// key == descending by value, ties to lower index (jax.lax.top_k order).
// ---------------------------------------------------------------------------
__global__ void uncert_keys_kernel(const float* __restrict__ outp,
                                   const float* __restrict__ resp,
                                   const float* __restrict__ og,
                                   unsigned long long* __restrict__ keys) {
  int t = blockIdx.x * blockDim.x + threadIdx.x;    // [0, BB*PAD)
  int b = t >> 16;
  int j = t & (PAD - 1);
  if (j >= KN) { keys[t] = 0ull; return; }

  float px = og[((size_t)b * KN + j) * 2 + 0];
  float py = og[((size_t)b * KN + j) * 2 + 1];
  float ix = px * (float)WW - 0.5f;
  float iy = py * (float)HH - 0.5f;
  float x0f = floorf(ix), y0f = floorf(iy);
  int   x0 = (int)x0f,    y0 = (int)y0f;
  float wx1 = ix - x0f, wx0 = 1.0f - wx1;
  float wy1 = iy - y0f, wy0 = 1.0f - wy1;

  const float* ob = outp + (size_t)b * NCLS * HW;
  const float* rb = resp + (size_t)b * NCLS * HW;

  float u = 0.0f;
#pragma unroll
  for (int cy = 0; cy < 2; ++cy) {
    int   yc = y0 + cy;
    float wy = cy ? wy1 : wy0;
    bool  vy = (yc >= 0) && (yc < HH);
    int   yi = iclamp(yc, 0, HH - 1);
#pragma unroll
    for (int cx = 0; cx < 2; ++cx) {
      int   xc = x0 + cx;
      float wx = cx ? wx1 : wx0;
      bool  vx = (xc >= 0) && (xc < WW);
      float w  = wy * wx * (float)(vx && vy);
      int   off = yi * WW + iclamp(xc, 0, WW - 1);
      float d, d0, d1, d2;
      d = ob[0 * HW + off] - rb[0 * HW + off]; d0 = d * d;
      d = ob[1 * HW + off] - rb[1 * HW + off]; d1 = d * d;
      d = ob[2 * HW + off] - rb[2 * HW + off]; d2 = d * d;
      float mx = fmaxf(d0, fmaxf(d1, d2));
      float mn = fminf(d0, fminf(d1, d2));
      float md = d0 + d1 + d2 - mx - mn;
      u += w * (md - mx);          // interp(top2) - interp(top1)
    }
  }
  keys[t] = ((unsigned long long)float_flip(u) << 32) |
            (unsigned long long)(0xFFFFFFFFu - (unsigned)j);
}

// ---------------------------------------------------------------------------
// Bitonic sort (descending), LDS-fused. Direction uses the GLOBAL index, so
// fused chunks implement exactly the flat bitonic network.
// ---------------------------------------------------------------------------
__launch_bounds__(SORT_T)
__global__ void bitonic_local_sort(unsigned long long* __restrict__ keys) {
  __shared__ unsigned long long s[CHUNK];           // 64 KB
  size_t base = (size_t)blockIdx.x * CHUNK;
  for (int i = threadIdx.x; i < CHUNK; i += SORT_T) s[i] = keys[base + i];
  __syncthreads();
  for (int k = 2; k <= CHUNK; k <<= 1) {
    for (int j = k >> 1; j > 0; j >>= 1) {
      for (int p = threadIdx.x; p < CHUNK / 2; p += SORT_T) {
        int i   = ((p & ~(j - 1)) << 1) | (p & (j - 1));
        int ixj = i | j;
        unsigned long long a = s[i], c = s[ixj];
        bool up = (((base + (size_t)i) & (size_t)k) == 0);
        bool sw = up ? (a < c) : (a > c);
        if (sw) { s[i] = c; s[ixj] = a; }
      }
      __syncthreads();
    }
  }
  for (int i = threadIdx.x; i < CHUNK; i += SORT_T) keys[base + i] = s[i];
}

__launch_bounds__(SORT_T)
__global__ void bitonic_local_merge(unsigned long long* __restrict__ keys,
                                    int kk) {
  __shared__ unsigned long long s[CHUNK];
  size_t base = (size_t)blockIdx.x * CHUNK;
  for (int i = threadIdx.x; i < CHUNK; i += SORT_T) s[i] = keys[base + i];
  __syncthreads();
  bool up = ((base & (size_t)kk) == 0);             // kk > CHUNK: uniform/block
  for (int j = CHUNK >> 1; j > 0; j >>= 1) {
    for (int p = threadIdx.x; p < CHUNK / 2; p += SORT_T) {
      int i   = ((p & ~(j - 1)) << 1) | (p & (j - 1));
      int ixj = i | j;
      unsigned long long a = s[i], c = s[ixj];
      bool sw = up ? (a < c) : (a > c);
      if (sw) { s[i] = c; s[ixj] = a; }
    }
    __syncthreads();
  }
  for (int i = threadIdx.x; i < CHUNK; i += SORT_T) keys[base + i] = s[i];
}

__global__ void bitonic_global_pass(unsigned long long* __restrict__ keys,
                                    int kk, int jj) {
  int t = blockIdx.x * blockDim.x + threadIdx.x;    // [0, BB*PAD)
  int b = t >> 16;
  int i = t & (PAD - 1);
  int ixj = i ^ jj;
  if (ixj <= i) return;
  unsigned long long* kb = keys + ((size_t)b << 16);
  unsigned long long a = kb[i], c = kb[ixj];
  bool up = ((i & kk) == 0);
  bool sw = up ? (a < c) : (a > c);
  if (sw) { kb[i] = c; kb[ixj] = a; }
}

// ---------------------------------------------------------------------------
// Kernel 3: fused gather + bilinear sampling + WMMA MLP + ReLU.
// One wave32 per 16-point tile; D(16x16) = Wpad(16x100) * Feat(100x16) via
// 25x V_WMMA_F32_16X16X4_F32. K reordered: [96 fine(x)][3 coarse(out)][1 zero]
// so the 24 fine steps are a branch-free pointer walk (channel-pair per lane
// half, +1-channel load folded into a 1MB immediate offset).
// ---------------------------------------------------------------------------
__global__ void render_wmma_kernel(const float* __restrict__ xp,
                                   const float* __restrict__ outp,
                                   const float* __restrict__ wmlp,
                                   const float* __restrict__ og,
                                   const float* __restrict__ cov,
                                   const unsigned long long* __restrict__ keys,
                                   float* __restrict__ rend) {
  // Padded weights in LDS: Wpad[m][k], m<16, k<100.
  // k in [0,96): fine  -> wmlp[m][3+k]
  // k in [96,99): coarse -> wmlp[m][k-96] ;  k==99 or m>=3 -> 0
  __shared__ float Wpad[16 * 100];
  for (int t = threadIdx.x; t < 16 * 100; t += blockDim.x) {
    int mm = t / 100, kk = t - mm * 100;
    float v = 0.0f;
    if (mm < NCLS) {
      if (kk < EMBED)          v = wmlp[mm * 99 + NCLS + kk];
      else if (kk < EMBED + 3) v = wmlp[mm * 99 + (kk - EMBED)];
    }
    Wpad[t] = v;
  }
  __syncthreads();

  int tid  = blockIdx.x * blockDim.x + threadIdx.x;
  int wave = tid >> 5;                 // 0 .. 4095
  int l    = tid & 31;
  int b    = wave >> 10;
  int n0   = (wave & 1023) << 4;
  int m    = l & 15;                   // A row == point-in-tile
  int hi   = l >> 4;                   // lane half -> K sub-pair
  int n    = n0 + m;

  // --- point coordinates (sorted importance order, then coverage) ---
  float px, py;
  if (n < NIMP) {
    unsigned long long key = keys[((size_t)b << 16) + n];
    unsigned j = 0xFFFFFFFFu - (unsigned)(key & 0xFFFFFFFFull);
    px = og[((size_t)b * KN + j) * 2 + 0];
    py = og[((size_t)b * KN + j) * 2 + 1];
  } else {
    int jc = n - NIMP;
    px = cov[((size_t)b * NCOV + jc) * 2 + 0];
    py = cov[((size_t)b * NCOV + jc) * 2 + 1];
  }

  // --- bilinear corner setup (zeros padding folded into weights) ---
  float ix = px * (float)WW - 0.5f;
  float iy = py * (float)HH - 0.5f;
  float x0f = floorf(ix), y0f = floorf(iy);
  int   x0 = (int)x0f,    y0 = (int)y0f;
  float wx1 = ix - x0f, wx0 = 1.0f - wx1;
  float wy1 = iy - y0f, wy0 = 1.0f - wy1;
  bool vx0 = (x0 >= 0) && (x0 < WW), vx1 = (x0 + 1 < WW);   // x0+1 >= 0 always when weight!=0
  bool vy0 = (y0 >= 0) && (y0 < HH), vy1 = (y0 + 1 < HH);
  int xi0 = iclamp(x0, 0, WW - 1), xi1 = iclamp(x0 + 1, 0, WW - 1);
  int yi0 = iclamp(y0, 0, HH - 1), yi1 = iclamp(y0 + 1, 0, HH - 1);
  float w00 = wx0 * wy0 * (float)(vx0 && vy0);
  float w01 = wx1 * wy0 * (float)(vx1 && vy0);
  float w10 = wx0 * wy1 * (float)(vx0 && vy1);
  float w11 = wx1 * wy1 * (float)(vx1 && vy1);
  int o00 = yi0 * WW + xi0, o01 = yi0 * WW + xi1;
  int o10 = yi1 * WW + xi0, o11 = yi1 * WW + xi1;

  // Persistent per-lane corner pointers into x, at this lane's channel 2*hi.
  const float* p00 = xp + (((size_t)(b * EMBED + 2 * hi)) << 18) + o00;
  const float* p01 = p00 + (o01 - o00);
  const float* p10 = p00 + (o10 - o00);
  const float* p11 = p00 + (o11 - o00);
  const int cstep = 4 << 18;           // 4 channel planes per K-step

  v8f acc = {};
  int wk = m * 100 + 2 * hi;           // LDS index of this lane's weight pair

  // ---- 24 fine K-steps (channels of x), branch-free ----
#pragma unroll 4
  for (int s = 0; s < 24; ++s) {
    v2f a, bv;
    a.x  = Wpad[wk];
    a.y  = Wpad[wk + 1];
    bv.x = w00 * p00[0]  + w01 * p01[0]  + w10 * p10[0]  + w11 * p11[0];
    bv.y = w00 * p00[HW] + w01 * p01[HW] + w10 * p10[HW] + w11 * p11[HW];
    acc = __builtin_amdgcn_wmma_f32_16x16x4_f32(
        false, a, false, bv, (short)0, acc, false, false);
    p00 += cstep; p01 += cstep; p10 += cstep; p11 += cstep;
    wk += 4;
  }

  // ---- coarse tail: K=96..99 -> out channels {2*hi, 2*hi+1}, ch 3 masked ----
  {
    int c0 = 2 * hi;                   // 0 or 2, always valid
    int c1 = 2 * hi + 1;               // 1 or 3(masked)
    const float* ob = outp + (((size_t)b * NCLS) << 18);
    const float* q0 = ob + ((size_t)c0 << 18);
    const float* q1 = ob + ((size_t)iclamp(c1, 0, 2) << 18);
    float mask1 = (c1 < NCLS) ? 1.0f : 0.0f;
    v2f a, bv;
    a.x  = Wpad[m * 100 + EMBED + 2 * hi];
    a.y  = Wpad[m * 100 + EMBED + 2 * hi + 1];      // Wpad[..][99] == 0
    bv.x = w00 * q0[o00] + w01 * q0[o01] + w10 * q0[o10] + w11 * q0[o11];
    bv.y = mask1 * (w00 * q1[o00] + w01 * q1[o01] + w10 * q1[o10] + w11 * q1[o11]);
    acc = __builtin_amdgcn_wmma_f32_16x16x4_f32(
        false, a, false, bv, (short)0, acc, false, false);
  }

  // D layout: VGPR r, lanes 0-15 -> row M=r (rows 0..2 are the classes)
  if (hi == 0) {
#pragma unroll
    for (int r = 0; r < NCLS; ++r) {
      rend[((size_t)b * NCLS + r) * NPTS + n] = fmaxf(acc[r], 0.0f);
    }
  }
}

// ---------------------------------------------------------------------------
extern "C" void kernel_launch(void* const* d_in, const int* in_sizes, int n_in,
                              void* d_out, int out_size, void* d_ws, size_t ws_size,
                              hipStream_t stream) {
  (void)in_sizes; (void)n_in; (void)out_size; (void)ws_size;
  const float* x   = (const float*)d_in[0];   // [4,96,512,512]
  const float* res = (const float*)d_in[1];   // [4,3,512,512]
  const float* out = (const float*)d_in[2];   // [4,3,512,512]
  const float* wm  = (const float*)d_in[3];   // [3,99]
  const float* og  = (const float*)d_in[4];   // [4,49152,2]
  const float* cov = (const float*)d_in[5];   // [4,4096,2]
  float* rend = (float*)d_out;                // [4,3,16384]
  unsigned long long* keys = (unsigned long long*)d_ws;   // 2 MB

  // 1) uncertainty keys (+ pad slots)
  uncert_keys_kernel<<<(BB * PAD) / 256, 256, 0, stream>>>(out, res, og, keys);

  // 2) bitonic sort, descending, per batch. LDS-fused: 10 launches total.
  const int nchunks = (BB * PAD) / CHUNK;           // 32
  bitonic_local_sort<<<nchunks, SORT_T, 0, stream>>>(keys);
  for (int k = CHUNK * 2; k <= PAD; k <<= 1) {
    for (int j = k >> 1; j >= CHUNK; j >>= 1)
      bitonic_global_pass<<<(BB * PAD) / 256, 256, 0, stream>>>(keys, k, j);
    bitonic_local_merge<<<nchunks, SORT_T, 0, stream>>>(keys, k);
  }

  // 3) fused gather + WMMA MLP + ReLU (one wave32 per 16-point tile)
  render_wmma_kernel<<<(BB * (NPTS / 16) * 32) / 256, 256, 0, stream>>>(
      x, out, wm, og, cov, keys, rend);
}